// MultiStepEvolveGCNH_85057532330353
// MI455X (gfx1250) — compile-verified
//
#include <hip/hip_runtime.h>
#include <hip/hip_bf16.h>
#include <math.h>

// Problem constants (match reference)
#define NODES   262144
#define NGR     16
#define NPERG   16384
#define EDGES   (NODES * 16)
#define TOPK    7
#define HIDF    64
#define INPF    32
#define STEPS   5
#define LNEPS   1e-5f

typedef float v2f __attribute__((ext_vector_type(2)));
typedef float v8f __attribute__((ext_vector_type(8)));

// ---------------------------------------------------------------------------
// zero fill (graph-capture safe; we must re-zero atomic targets every call)
// ---------------------------------------------------------------------------
__global__ void k_fill_zero(float* __restrict__ p, size_t n) {
    size_t i = (size_t)blockIdx.x * blockDim.x + threadIdx.x;
    size_t stride = (size_t)gridDim.x * blockDim.x;
    for (; i < n; i += stride) p[i] = 0.0f;
}

// ---------------------------------------------------------------------------
// per-node score = h[n,:] . (p/||p||)
// ---------------------------------------------------------------------------
__global__ void k_scores(const float* __restrict__ h, const float* __restrict__ p,
                         float* __restrict__ scores, int din) {
    int n = blockIdx.x * blockDim.x + threadIdx.x;
    if (n >= NODES) return;
    float pn = 0.f;
    for (int k = 0; k < din; ++k) pn += p[k] * p[k];
    float inv = rsqrtf(pn);
    const float* row = h + (size_t)n * din;
    float s = 0.f;
    for (int k = 0; k < din; ++k) s += row[k] * p[k];
    scores[n] = s * inv;
}

// ---------------------------------------------------------------------------
// per-graph top-7 + weighted mean summarization: Z[g,:] (one block per graph)
// ---------------------------------------------------------------------------
__global__ void k_topk_summarize(const float* __restrict__ h,
                                 const float* __restrict__ scores,
                                 float* __restrict__ Z, int din) {
    __shared__ float red_v[256];
    __shared__ int   red_i[256];
    __shared__ int   chosen[TOPK];
    __shared__ float chosenW[TOPK];
    const int g   = blockIdx.x;
    const int tid = threadIdx.x;
    const int base = g * NPERG;

    for (int k = 0; k < TOPK; ++k) {
        float bv = -3.4e38f; int bi = 0;
        for (int i = tid; i < NPERG; i += 256) {
            bool taken = false;
            for (int c = 0; c < k; ++c) if (chosen[c] == i) taken = true;
            float s = scores[base + i];
            if (!taken && s > bv) { bv = s; bi = i; }
        }
        red_v[tid] = bv; red_i[tid] = bi;
        __syncthreads();
        for (int off = 128; off > 0; off >>= 1) {
            if (tid < off && red_v[tid + off] > red_v[tid]) {
                red_v[tid] = red_v[tid + off]; red_i[tid] = red_i[tid + off];
            }
            __syncthreads();
        }
        if (tid == 0) { chosen[k] = red_i[0]; chosenW[k] = tanhf(red_v[0]); }
        __syncthreads();
    }
    for (int f = tid; f < din; f += 256) {
        float acc = 0.f;
        for (int k = 0; k < TOPK; ++k)
            acc += chosenW[k] * h[(size_t)(base + chosen[k]) * din + f];
        Z[g * din + f] = acc * (1.0f / (float)TOPK);
    }
}

// ---------------------------------------------------------------------------
// GRU cell + hidden-mean -> evolved GCN weight W[P] (P = din*64).
// One thread per hidden column j, all 16 batch rows accumulated in registers,
// 16xP hidden state streamed through LDS in chunks so Whh is read exactly once.
// ---------------------------------------------------------------------------
#define GRU_CH 512
__global__ void k_gru_evolve(const float* __restrict__ Z,
                             const float* __restrict__ gh,
                             const float* __restrict__ Wih,
                             const float* __restrict__ Whh,
                             const float* __restrict__ bih,
                             const float* __restrict__ bhh,
                             float* __restrict__ Wout, int din, int P) {
    __shared__ float sZ[16 * 64];
    __shared__ float sGH[16 * GRU_CH];
    const int tid = threadIdx.x;
    const int j = blockIdx.x * blockDim.x + tid;   // j in [0, P)

    for (int i = tid; i < 16 * din; i += blockDim.x) sZ[i] = Z[i];
    __syncthreads();

    float r_acc[16], z_acc[16], in_acc[16], hn_acc[16];
    const float br  = bih[j]         + bhh[j];
    const float bz  = bih[P + j]     + bhh[P + j];
    const float bin = bih[2 * P + j];
    const float bhn = bhh[2 * P + j];
#pragma unroll
    for (int b = 0; b < 16; ++b) { r_acc[b] = br; z_acc[b] = bz; in_acc[b] = bin; hn_acc[b] = bhn; }

    // input contribution: Z @ Wih^T
    const float* wr = Wih + (size_t)j * din;
    const float* wz = Wih + (size_t)(P + j) * din;
    const float* wn = Wih + (size_t)(2 * P + j) * din;
    for (int k = 0; k < din; ++k) {
        float a = wr[k], c = wz[k], d = wn[k];
#pragma unroll
        for (int b = 0; b < 16; ++b) {
            float zz = sZ[b * din + k];
            r_acc[b] += a * zz; z_acc[b] += c * zz; in_acc[b] += d * zz;
        }
    }
    // hidden contribution: gh @ Whh^T, chunked through LDS
    const float* hr = Whh + (size_t)j * P;
    const float* hz = Whh + (size_t)(P + j) * P;
    const float* hn = Whh + (size_t)(2 * P + j) * P;
    for (int c0 = 0; c0 < P; c0 += GRU_CH) {
        __syncthreads();
        for (int i = tid; i < 16 * GRU_CH; i += blockDim.x) {
            int b = i / GRU_CH, k = i % GRU_CH;
            sGH[i] = gh[(size_t)b * P + c0 + k];
        }
        __syncthreads();
        if (c0 + GRU_CH < P) {
            __builtin_prefetch(hr + c0 + GRU_CH, 0, 1);
            __builtin_prefetch(hz + c0 + GRU_CH, 0, 1);
            __builtin_prefetch(hn + c0 + GRU_CH, 0, 1);
        }
        for (int k = 0; k < GRU_CH; ++k) {
            float a = hr[c0 + k], c = hz[c0 + k], d = hn[c0 + k];
#pragma unroll
            for (int b = 0; b < 16; ++b) {
                float hh = sGH[b * GRU_CH + k];
                r_acc[b] += a * hh; z_acc[b] += c * hh; hn_acc[b] += d * hh;
            }
        }
    }
    // gates + mean over batch -> evolved weight element
    float wsum = 0.f;
#pragma unroll
    for (int b = 0; b < 16; ++b) {
        float r  = 1.f / (1.f + expf(-r_acc[b]));
        float zg = 1.f / (1.f + expf(-z_acc[b]));
        float n  = tanhf(in_acc[b] + r * hn_acc[b]);
        float hp = gh[(size_t)b * P + j];
        wsum += (1.f - zg) * n + zg * hp;
    }
    Wout[j] = wsum * (1.0f / 16.0f);
}

// ---------------------------------------------------------------------------
// xw = h @ W + bias, via V_WMMA_F32_16X16X4_F32.
// One wave32 per 16-node row block; 4 accumulator tiles cover 64 output cols.
// A layout: lanes 0-15 hold {K0,K1}, lanes 16-31 hold {K2,K3}, M = lane%16.
// B layout mirrored with N in lanes. C/D: 8 VGPRs, M = vgpr (+8 upper half).
// ---------------------------------------------------------------------------
template <int DIN>
__global__ void k_xw_wmma(const float* __restrict__ h, const float* __restrict__ W,
                          const float* __restrict__ bias, float* __restrict__ xw) {
    const int wave  = blockIdx.x * (blockDim.x >> 5) + (threadIdx.x >> 5);
    const int lane  = threadIdx.x & 31;
    const int mrow  = lane & 15;
    const int khalf = lane >> 4;          // 0 -> K0/K1, 1 -> K2/K3
    const size_t row0 = (size_t)wave * 16;

    v8f acc[4] = {v8f{}, v8f{}, v8f{}, v8f{}};
#pragma unroll
    for (int k = 0; k < DIN; k += 4) {
        const float* ap = h + (row0 + mrow) * DIN + k + khalf * 2;
        v2f a; a.x = ap[0]; a.y = ap[1];
#pragma unroll
        for (int nt = 0; nt < 4; ++nt) {
            const float* bpp = W + (size_t)(k + khalf * 2) * HIDF + nt * 16 + mrow;
            v2f b; b.x = bpp[0]; b.y = bpp[HIDF];
            acc[nt] = __builtin_amdgcn_wmma_f32_16x16x4_f32(
                false, a, false, b, (short)0, acc[nt], false, false);
        }
    }
#pragma unroll
    for (int nt = 0; nt < 4; ++nt) {
        const int col = nt * 16 + mrow;
        const float bv = bias[col];
#pragma unroll
        for (int v = 0; v < 8; ++v) {
            size_t node = row0 + v + khalf * 8;
            xw[node * HIDF + col] = acc[nt][v] + bv;
        }
    }
}

// ---------------------------------------------------------------------------
// degree: deg[dst] += ew  (thread per edge)
// ---------------------------------------------------------------------------
__global__ void k_degree(const int* __restrict__ dst, const float* __restrict__ ew,
                         float* __restrict__ deg) {
    size_t e = (size_t)blockIdx.x * blockDim.x + threadIdx.x;
    if (e < (size_t)EDGES) atomicAdd(&deg[dst[e]], ew[e]);
}

// ---------------------------------------------------------------------------
// edge aggregation: agg[dst,:] += norm * xw[src,:]  (64 threads per edge)
// xw/agg are 67 MB each -> L2 resident on the 192 MB L2.
// ---------------------------------------------------------------------------
__global__ void k_edge_aggregate(const float* __restrict__ xw,
                                 const int* __restrict__ src,
                                 const int* __restrict__ dst,
                                 const float* __restrict__ ew,
                                 const float* __restrict__ degsum,
                                 float* __restrict__ agg) {
    size_t idx = (size_t)blockIdx.x * blockDim.x + threadIdx.x;
    size_t e = idx >> 6;
    int f = (int)(idx & 63);
    int s = src[e], d = dst[e];
    float norm = ew[e] * rsqrtf((degsum[s] + 1.f) * (degsum[d] + 1.f));
    atomicAdd(&agg[(size_t)d * HIDF + f], norm * xw[(size_t)s * HIDF + f]);
}

// ---------------------------------------------------------------------------
// self-loop + LayerNorm + ReLU, one wave32 per node (2 features per lane)
// ---------------------------------------------------------------------------
__global__ void k_ln_relu(const float* __restrict__ agg, const float* __restrict__ xw,
                          const float* __restrict__ degsum,
                          const float* __restrict__ g, const float* __restrict__ b,
                          float* __restrict__ hout) {
    const int node = blockIdx.x * (blockDim.x >> 5) + (threadIdx.x >> 5);
    const int lane = threadIdx.x & 31;
    const size_t base = (size_t)node * HIDF;
    const float invdeg = 1.f / (degsum[node] + 1.f);
    float v0 = agg[base + lane]      + xw[base + lane]      * invdeg;
    float v1 = agg[base + 32 + lane] + xw[base + 32 + lane] * invdeg;
    float s = v0 + v1;
#pragma unroll
    for (int off = 16; off > 0; off >>= 1) s += __shfl_xor(s, off, 32);
    const float mu = s * (1.0f / HIDF);
    float d0 = v0 - mu, d1 = v1 - mu;
    float vs = d0 * d0 + d1 * d1;
#pragma unroll
    for (int off = 16; off > 0; off >>= 1) vs += __shfl_xor(vs, off, 32);
    const float rstd = rsqrtf(vs * (1.0f / HIDF) + LNEPS);
    float o0 = d0 * rstd * g[lane]      + b[lane];
    float o1 = d1 * rstd * g[32 + lane] + b[32 + lane];
    hout[base + lane]      = o0 > 0.f ? o0 : 0.f;
    hout[base + 32 + lane] = o1 > 0.f ? o1 : 0.f;
}

// ---------------------------------------------------------------------------
// stacked prediction heads: out[n,s,o] = h[n,:] @ Wp[s,:,o] + bp[s,o]
// ---------------------------------------------------------------------------
__global__ void k_predict(const float* __restrict__ h, const float* __restrict__ Wp,
                          const float* __restrict__ bp, float* __restrict__ out) {
    int n = blockIdx.x * blockDim.x + threadIdx.x;
    if (n >= NODES) return;
    const float* row = h + (size_t)n * HIDF;
#pragma unroll
    for (int s = 0; s < STEPS; ++s) {
        float a0 = bp[s * 2], a1 = bp[s * 2 + 1];
        for (int d = 0; d < HIDF; ++d) {
            float x = row[d];
            a0 += x * Wp[(s * HIDF + d) * 2 + 0];
            a1 += x * Wp[(s * HIDF + d) * 2 + 1];
        }
        out[((size_t)n * STEPS + s) * 2 + 0] = a0;
        out[((size_t)n * STEPS + s) * 2 + 1] = a1;
    }
}

// ---------------------------------------------------------------------------
extern "C" void kernel_launch(void* const* d_in, const int* in_sizes, int n_in,
                              void* d_out, int out_size, void* d_ws, size_t ws_size,
                              hipStream_t stream) {
    (void)in_sizes; (void)n_in; (void)out_size; (void)ws_size;
    const float* x     = (const float*)d_in[0];
    const int*   eidx  = (const int*)d_in[1];
    const float* ew    = (const float*)d_in[2];
    const float* p0    = (const float*)d_in[5];
    const float* p1    = (const float*)d_in[6];
    const float* Wih0  = (const float*)d_in[7];
    const float* Whh0  = (const float*)d_in[8];
    const float* bih0  = (const float*)d_in[9];
    const float* bhh0  = (const float*)d_in[10];
    const float* Wih1  = (const float*)d_in[11];
    const float* Whh1  = (const float*)d_in[12];
    const float* bih1  = (const float*)d_in[13];
    const float* bhh1  = (const float*)d_in[14];
    const float* gruh0 = (const float*)d_in[15];
    const float* gruh1 = (const float*)d_in[16];
    const float* gcnb0 = (const float*)d_in[17];
    const float* gcnb1 = (const float*)d_in[18];
    const float* lng0  = (const float*)d_in[19];
    const float* lnb0  = (const float*)d_in[20];
    const float* lng1  = (const float*)d_in[21];
    const float* lnb1  = (const float*)d_in[22];
    const float* Wp    = (const float*)d_in[23];
    const float* bp    = (const float*)d_in[24];
    const int* src = eidx;
    const int* dst = eidx + EDGES;

    const int P0 = INPF * HIDF;   // 2048
    const int P1 = HIDF * HIDF;   // 4096

    // workspace layout (floats)
    float* ws     = (float*)d_ws;
    float* scores = ws;                                    // NODES
    float* deg    = scores + NODES;                        // NODES
    float* Z      = deg + NODES;                           // 16*64
    float* Wmat   = Z + 16 * 64;                           // max P1
    float* bufA   = Wmat + P1;                             // NODES*64  (xw)
    float* bufB   = bufA + (size_t)NODES * HIDF;           // NODES*64  (agg)
    float* bufH   = bufB + (size_t)NODES * HIDF;           // NODES*64  (h)

    // degree (edge-only, shared by both layers)
    k_fill_zero<<<2048, 256, 0, stream>>>(deg, (size_t)NODES);
    k_degree<<<EDGES / 256, 256, 0, stream>>>(dst, ew, deg);

    // ---------------- layer 0 (din = 32) ----------------
    k_scores<<<NODES / 256, 256, 0, stream>>>(x, p0, scores, INPF);
    k_topk_summarize<<<NGR, 256, 0, stream>>>(x, scores, Z, INPF);
    k_gru_evolve<<<P0 / 256, 256, 0, stream>>>(Z, gruh0, Wih0, Whh0, bih0, bhh0,
                                               Wmat, INPF, P0);
    k_xw_wmma<INPF><<<NODES / (16 * 8), 256, 0, stream>>>(x, Wmat, gcnb0, bufA);
    k_fill_zero<<<8192, 256, 0, stream>>>(bufB, (size_t)NODES * HIDF);
    k_edge_aggregate<<<(size_t)EDGES * HIDF / 256, 256, 0, stream>>>(
        bufA, src, dst, ew, deg, bufB);
    k_ln_relu<<<NODES / 8, 256, 0, stream>>>(bufB, bufA, deg, lng0, lnb0, bufH);

    // ---------------- layer 1 (din = 64) ----------------
    k_scores<<<NODES / 256, 256, 0, stream>>>(bufH, p1, scores, HIDF);
    k_topk_summarize<<<NGR, 256, 0, stream>>>(bufH, scores, Z, HIDF);
    k_gru_evolve<<<P1 / 256, 256, 0, stream>>>(Z, gruh1, Wih1, Whh1, bih1, bhh1,
                                               Wmat, HIDF, P1);
    k_xw_wmma<HIDF><<<NODES / (16 * 8), 256, 0, stream>>>(bufH, Wmat, gcnb1, bufA);
    k_fill_zero<<<8192, 256, 0, stream>>>(bufB, (size_t)NODES * HIDF);
    k_edge_aggregate<<<(size_t)EDGES * HIDF / 256, 256, 0, stream>>>(
        bufA, src, dst, ew, deg, bufB);
    k_ln_relu<<<NODES / 8, 256, 0, stream>>>(bufB, bufA, deg, lng1, lnb1, bufH);

    // ---------------- prediction heads ----------------
    k_predict<<<NODES / 256, 256, 0, stream>>>(bufH, Wp, bp, (float*)d_out);
}